// GroupGCN_45861660786780
// MI455X (gfx1250) — compile-verified
//
#include <hip/hip_runtime.h>
#include <math.h>

typedef __attribute__((ext_vector_type(2))) float v2f;
typedef __attribute__((ext_vector_type(8))) float v8f;

// ---------------------------------------------------------------------------
// Degree / normalization
// ---------------------------------------------------------------------------
__global__ void fill_zero_f32(float* __restrict__ p, int n) {
    int i = blockIdx.x * blockDim.x + threadIdx.x;
    if (i < n) p[i] = 0.0f;
}

__global__ void deg_count(const int* __restrict__ ei, float* __restrict__ deg, int E) {
    int e = blockIdx.x * blockDim.x + threadIdx.x;
    if (e < E) atomicAdd(&deg[ei[E + e]], 1.0f);   // dst column
}

__global__ void make_dinv(float* __restrict__ d, int n) {
    int i = blockIdx.x * blockDim.x + threadIdx.x;
    if (i < n) d[i] = rsqrtf(d[i] + 1.0f);         // +1 = self loop, deg>0 always
}

// ---------------------------------------------------------------------------
// Dense GEMM via fp32 WMMA: H[M,OUT] = A[M,K] * W[K,OUT]
// One wave per 16x16 output tile, K stepped by 4 (V_WMMA_F32_16X16X4_F32).
// A-matrix 16x4 f32 layout: lanes 0-15 -> (K=0,K=1); lanes 16-31 -> (K=2,K=3).
// C/D layout: VGPR r, lanes 0-15 -> M=r, lanes 16-31 -> M=8+r; N = lane&15.
// ---------------------------------------------------------------------------
template <int K, int OUT>
__global__ void gemm_wmma(const float* __restrict__ A, const float* __restrict__ W,
                          float* __restrict__ H, int M) {
    const int lane = threadIdx.x & 31;
    const int wave = threadIdx.x >> 5;
    constexpr int NT = (OUT + 15) / 16;
    int tile   = blockIdx.x * (blockDim.x >> 5) + wave;
    int mtiles = (M + 15) >> 4;
    int mt = tile / NT;
    int nt = tile - mt * NT;
    if (mt >= mtiles) return;                       // wave-uniform

    const int half = lane >> 4;
    const int l15  = lane & 15;

    int row  = mt * 16 + l15;
    int arow = row < M ? row : (M - 1);             // clamp: garbage rows never stored
    int col  = nt * 16 + l15;
    int cidx = col < OUT ? col : 0;                 // clamp: garbage cols never stored

    const float* ap = A + (size_t)arow * K;
    v8f acc = {};
#pragma unroll
    for (int k0 = 0; k0 < K; k0 += 4) {
        const int kk = k0 + 2 * half;
        v2f a = *(const v2f*)(ap + kk);             // K multiple of 4, kk even -> 8B aligned
        v2f b;
        b.x = W[kk * OUT + cidx];
        b.y = W[(kk + 1) * OUT + cidx];
        acc = __builtin_amdgcn_wmma_f32_16x16x4_f32(
            /*neg_a=*/false, a, /*neg_b=*/false, b,
            /*c_mod=*/(short)0, acc, /*reuse_a=*/false, /*reuse_b=*/false);
    }
#pragma unroll
    for (int r = 0; r < 8; ++r) {
        int m = mt * 16 + r + 8 * half;
        if (m < M && col < OUT) H[(size_t)m * OUT + col] = acc[r];
    }
}

// ---------------------------------------------------------------------------
// Aggregation: agg = self-loop init + edge scatter (L2-resident atomics)
// ---------------------------------------------------------------------------
__global__ void init_agg(const float* __restrict__ h, const float* __restrict__ dinv,
                         float* __restrict__ agg, int total, int l2o) {
    int i = blockIdx.x * blockDim.x + threadIdx.x;
    if (i < total) {
        float di = dinv[i >> l2o];
        agg[i] = h[i] * di * di;                    // self-loop contribution
    }
}

__global__ void edge_scatter(const float* __restrict__ h, const int* __restrict__ ei,
                             const float* __restrict__ dinv, float* __restrict__ agg,
                             int E, int l2o) {
    unsigned i = blockIdx.x * blockDim.x + threadIdx.x;
    unsigned e = i >> l2o;
    if (e < (unsigned)E) {
        int c = i & ((1 << l2o) - 1);
        int s = ei[e];
        int d = ei[E + e];
        float nrm = dinv[s] * dinv[d];
        atomicAdd(&agg[((size_t)d << l2o) + c], h[((size_t)s << l2o) + c] * nrm);
    }
}

__global__ void bias_relu(float* __restrict__ agg, const float* __restrict__ b,
                          int total, int l2o) {
    int i = blockIdx.x * blockDim.x + threadIdx.x;
    if (i < total) {
        float v = agg[i] + b[i & ((1 << l2o) - 1)];
        agg[i] = v > 0.0f ? v : 0.0f;
    }
}

__global__ void final_lsm(const float* __restrict__ agg, const float* __restrict__ b,
                          float* __restrict__ out, int n) {
    int i = blockIdx.x * blockDim.x + threadIdx.x;
    if (i < n) {
        float4 v = *(const float4*)(agg + (size_t)i * 4);
        v.x += b[0]; v.y += b[1]; v.z += b[2]; v.w += b[3];
        float m  = fmaxf(fmaxf(v.x, v.y), fmaxf(v.z, v.w));
        float s  = expf(v.x - m) + expf(v.y - m) + expf(v.z - m) + expf(v.w - m);
        float ls = m + logf(s);
        float4 o = make_float4(v.x - ls, v.y - ls, v.z - ls, v.w - ls);
        *(float4*)(out + (size_t)i * 4) = o;
    }
}

// ---------------------------------------------------------------------------
// Host launcher
// ---------------------------------------------------------------------------
extern "C" void kernel_launch(void* const* d_in, const int* in_sizes, int n_in,
                              void* d_out, int out_size, void* d_ws, size_t ws_size,
                              hipStream_t stream) {
    (void)n_in; (void)out_size; (void)ws_size;

    const float* x  = (const float*)d_in[0];          // [N, 256]
    const int*   ei = (const int*)d_in[1];            // [2, E]  (JAX default: int32)
    const float* W[4] = { (const float*)d_in[2], (const float*)d_in[4],
                          (const float*)d_in[6], (const float*)d_in[8] };
    const float* b[4] = { (const float*)d_in[3], (const float*)d_in[5],
                          (const float*)d_in[7], (const float*)d_in[9] };
    float* out = (float*)d_out;

    const int N = in_sizes[0] / 256;
    const int E = in_sizes[1] / 2;

    char*  ws   = (char*)d_ws;
    float* dinv = (float*)ws;                              // N floats
    float* buf0 = (float*)(ws + (1u << 20));               // h   (<= N*64 f32 = 25.6MB)
    float* buf1 = (float*)(ws + (1u << 20) + (27u << 20)); // agg/act (<= 25.6MB)

    const int TB = 256;

    // Degree + D^{-1/2} (computed once, reused by every layer)
    fill_zero_f32<<<(N + TB - 1) / TB, TB, 0, stream>>>(dinv, N);
    deg_count   <<<(E + TB - 1) / TB, TB, 0, stream>>>(ei, dinv, E);
    make_dinv   <<<(N + TB - 1) / TB, TB, 0, stream>>>(dinv, N);

#define LAYER(KK, OO, L2O, ACT, WW, BB, DO_RELU)                                   \
    {                                                                              \
        int mtiles = (N + 15) >> 4;                                                \
        int ntiles = ((OO) + 15) / 16;                                             \
        int tiles  = mtiles * ntiles;                                              \
        gemm_wmma<KK, OO><<<(tiles + 7) / 8, 256, 0, stream>>>(ACT, WW, buf0, N);  \
        int total = N * (OO);                                                      \
        init_agg<<<(total + TB - 1) / TB, TB, 0, stream>>>(buf0, dinv, buf1,       \
                                                           total, L2O);            \
        unsigned ew = (unsigned)E << (L2O);                                        \
        edge_scatter<<<(ew + TB - 1) / TB, TB, 0, stream>>>(buf0, ei, dinv, buf1,  \
                                                            E, L2O);               \
        if (DO_RELU)                                                               \
            bias_relu<<<(total + TB - 1) / TB, TB, 0, stream>>>(buf1, BB, total,   \
                                                                L2O);              \
    }

    // act for layer l>1 lives in buf1; GEMM fully consumes it before init_agg
    // overwrites buf1 (single-stream ordering makes this safe).
    LAYER(256, 64, 6, x,    W[0], b[0], true)
    LAYER(64,  32, 5, buf1, W[1], b[1], true)
    LAYER(32,  16, 4, buf1, W[2], b[2], true)
    LAYER(16,  4,  2, buf1, W[3], b[3], false)
#undef LAYER

    final_lsm<<<(N + TB - 1) / TB, TB, 0, stream>>>(buf1, b[3], out, N);
}